// SE3MovingFrameInvariants_42511586296170
// MI455X (gfx1250) — compile-verified
//
#include <hip/hip_runtime.h>
#include <math.h>

// ---------------------------------------------------------------------------
// f is [1,64,64,64,8] f32; memory idx = x*32768 + y*512 + z*8 + c.
// 19 separable conv passes (WMMA band-matrix matmuls), then per-voxel einsums.
// ---------------------------------------------------------------------------
#define NX 64
#define NY 64
#define NZ 64
#define NC 8
#define NFIELD ((size_t)NX * NY * NZ * NC)   // 2,097,152 floats per field
#define NLINES 32768                          // lines per conv pass

typedef __attribute__((ext_vector_type(2))) float v2f;
typedef __attribute__((ext_vector_type(4))) float v4f;
typedef __attribute__((ext_vector_type(8))) float v8f;

struct Taps { float g[9]; };

// ---------------------------------------------------------------------------
// Conv along one axis as a shared-operand matmul:
//   Out(64 x N) = W(64 x 64) * In(64 x N),  W[m,k] = g[k-m+4] on the band.
// One wave handles a 16-line tile: 64x16 In tile cached in 32 VGPRs,
// 4 M-blocks x 16 K-steps of V_WMMA_F32_16X16X4_F32 (fp32-exact).
// Strides are template parameters so every access is base + immediate offset
// (all offsets < 2^23 bytes -> fit the 24-bit IOFFSET field).
// ---------------------------------------------------------------------------
template <int S0, int SH, int SM>
__global__ __launch_bounds__(32)
void conv_wmma(const float* __restrict__ in, float* __restrict__ out, Taps taps)
{
    const int lane = threadIdx.x;
    const int hi   = lane >> 4;          // 0: lanes 0-15, 1: lanes 16-31
    const int lo   = lane & 15;          // M (for A/D) and N (for B/D)
    const int l    = blockIdx.x * 16 + lo;                 // this lane's line
    const int base = (l >> 9) * SH + ((l >> 3) & 63) * SM + (l & 7);
    const float* __restrict__ pin  = in  + base;
    float* __restrict__       pout = out + base;

    // Load the 64x16 In tile.  B layout (4x16 f32): VGPR0 = K0 (lanes 0-15),
    // K2 (lanes 16-31); VGPR1 = K1 / K3; N = lane&15.
    const int kh = 2 * hi;               // this half-wave's K offset
    v2f b[16];
#pragma unroll
    for (int ks = 0; ks < 16; ++ks) {
        b[ks].x = pin[(4 * ks + kh    ) * S0];
        b[ks].y = pin[(4 * ks + kh + 1) * S0];
    }

#pragma unroll
    for (int mb = 0; mb < 4; ++mb) {
        const int m = mb * 16 + lo;      // A-matrix row M = lane&15
        v8f acc = {};
#pragma unroll
        for (int ks = 0; ks < 16; ++ks) {
            // Band-matrix A tile from the 9 taps (A layout: VGPR0 = K0
            // lanes0-15 / K2 lanes16-31; VGPR1 = K1 / K3; M = lane&15).
            const int d0 = 4 * ks + kh - m;
            const int d1 = d0 + 1;
            v2f a;
            a.x = (d0 >= -4 && d0 <= 4) ? taps.g[d0 + 4] : 0.0f;
            a.y = (d1 >= -4 && d1 <= 4) ? taps.g[d1 + 4] : 0.0f;
            acc = __builtin_amdgcn_wmma_f32_16x16x4_f32(
                      false, a, false, b[ks], (short)0, acc, false, false);
        }
        // D layout: VGPR r -> M = mb*16 + r + 8*hi, N = lane&15 (== base line)
#pragma unroll
        for (int r = 0; r < 8; ++r)
            pout[(mb * 16 + r + 8 * hi) * S0] = acc[r];
    }
}

// ---------------------------------------------------------------------------
// Final fused stage: per (voxel, channel) thread gathers the 10 derivative
// fields (regular loads -> L2-resident), applies the per-voxel 3x3 / 6x6
// transforms (unique matrices per voxel -> no reuse -> VALU FMA), and writes
// the 80-channel output.  frame/coefs/out are touched exactly once -> use
// non-temporal policy so they do not evict the ws fields from the 192MB L2.
// ---------------------------------------------------------------------------
__global__ __launch_bounds__(256)
void fuse_kernel(const float* __restrict__ ws, const float* __restrict__ frame,
                 const float* __restrict__ coefs, float* __restrict__ out)
{
    const int t = blockIdx.x * 256 + threadIdx.x;   // (voxel, channel)
    const int v = t >> 3;
    const int c = t & 7;

    const int slots[10] = {0, 1, 2, 9, 10, 11, 12, 13, 14, 15};
    float d[10];
#pragma unroll
    for (int i = 0; i < 10; ++i) d[i] = ws[(size_t)slots[i] * NFIELD + t];

    float* o = out + (size_t)v * 80;
    __builtin_nontemporal_store(d[0], &o[c]);        // u

    const float* F = frame + (size_t)t * 9;          // [3][3], 4B-aligned only
    float fr[9];
#pragma unroll
    for (int i = 0; i < 9; ++i) fr[i] = __builtin_nontemporal_load(&F[i]);
#pragma unroll
    for (int k = 0; k < 3; ++k) {
        float s = d[1] * fr[0 * 3 + k];
        s = fmaf(d[2], fr[1 * 3 + k], s);
        s = fmaf(d[3], fr[2 * 3 + k], s);
        __builtin_nontemporal_store(s, &o[8 + c * 3 + k]);
    }

    // coefs row = 36 floats = 144 B, 16B-aligned -> 9 x b128 NT loads.
    const v4f* Cq = (const v4f*)(coefs + (size_t)t * 36);
    float cf[36];
#pragma unroll
    for (int i = 0; i < 9; ++i) {
        v4f q = __builtin_nontemporal_load(&Cq[i]);
        cf[4 * i + 0] = q.x; cf[4 * i + 1] = q.y;
        cf[4 * i + 2] = q.z; cf[4 * i + 3] = q.w;
    }
#pragma unroll
    for (int k = 0; k < 6; ++k) {
        float s = 0.0f;
#pragma unroll
        for (int dd = 0; dd < 6; ++dd) s = fmaf(d[4 + dd], cf[dd * 6 + k], s);
        __builtin_nontemporal_store(s, &o[32 + c * 6 + k]);
    }
}

// ---------------------------------------------------------------------------
// Host side
// ---------------------------------------------------------------------------
static void make_taps(Taps t[3])
{
    const float sigma = 1.5f;
    const float pi = 3.14159265358979323846f;
    for (int i = 0; i < 9; ++i) {
        const float x  = (float)(i - 4);
        const float g0 = expf(-x * x / (2.0f * sigma * sigma)) /
                         sqrtf(2.0f * pi * sigma);
        const float tt = x / sigma;
        t[0].g[i] = g0;                                      // order 0
        t[1].g[i] = -tt * g0 / sigma;                        // order 1
        t[2].g[i] = (tt * tt - 1.0f) * g0 / (sigma * sigma); // order 2
    }
}

extern "C" void kernel_launch(void* const* d_in, const int* in_sizes, int n_in,
                              void* d_out, int out_size, void* d_ws, size_t ws_size,
                              hipStream_t stream)
{
    (void)in_sizes; (void)n_in; (void)out_size; (void)ws_size;
    const float* f     = (const float*)d_in[0];
    const float* frame = (const float*)d_in[1];
    const float* coefs = (const float*)d_in[2];
    float*       out   = (float*)d_out;
    float*       ws    = (float*)d_ws;   // 16 fields x 8 MiB = 128 MiB

    Taps g[3];
    make_taps(g);

    auto slot = [&](int s) { return ws + (size_t)s * NFIELD; };

    const dim3 grid(NLINES / 16), blk(32);

    // Pass 1: conv along X (axis stride 32768; lines over (y,z,c)).
    // fx[a] -> slots 0..2
    for (int a = 0; a < 3; ++a)
        conv_wmma<32768, 512, 8><<<grid, blk, 0, stream>>>(f, slot(a), g[a]);

    // Pass 2: conv along Y (axis stride 512; lines over (x,z,c)).
    // fxy slots: (0,0)->3 (1,0)->4 (0,1)->5 (2,0)->6 (1,1)->7 (0,2)->8
    struct P { int in, k, out; };
    const P p2[6] = {{0,0,3},{1,0,4},{0,1,5},{2,0,6},{1,1,7},{0,2,8}};
    for (int i = 0; i < 6; ++i)
        conv_wmma<512, 32768, 8><<<grid, blk, 0, stream>>>(
            slot(p2[i].in), slot(p2[i].out), g[p2[i].k]);

    // Pass 3: conv along Z (axis stride 8; lines over (x,y,c)).
    // combos: (0,0,0)(1,0,0)(0,1,0)(0,0,1)(2,0,0)(1,1,0)(0,2,0)(1,0,1)(0,1,1)(0,0,2)
    // derivs d0..d2 -> slots 0..2 (recycle fx), d3..d9 -> slots 9..15.
    const P p3[10] = {{3,0,0},{4,0,1},{5,0,2},{3,1,9},{6,0,10},
                      {7,0,11},{8,0,12},{4,1,13},{5,1,14},{3,2,15}};
    for (int i = 0; i < 10; ++i)
        conv_wmma<8, 32768, 512><<<grid, blk, 0, stream>>>(
            slot(p3[i].in), slot(p3[i].out), g[p3[i].k]);

    // Fused einsums + output assembly (bandwidth-dominant stage).
    fuse_kernel<<<(int)(NFIELD / 256), 256, 0, stream>>>(ws, frame, coefs, out);
}